// InferenceBox_28123445854516
// MI455X (gfx1250) — compile-verified
//
#include <hip/hip_runtime.h>
#include <math.h>

// ---------------- problem constants (from reference) ----------------
#define BATCH       64
#define NBOX        24564
#define ROW         85          // 4 locs + 81 class logits
#define NCLS        80          // background dropped
#define TOPK        200
#define KPAD        208         // 13 WMMA tiles of 16
#define NTILE       13
#define CONF_THR    0.01f
#define IOU_THR     0.45f
#define THREADS     256
#define NPAD        24576       // NBOX rounded up
#define CANDCAP     512         // candidate buffer for sort (>= TOPK + ties)

typedef __attribute__((ext_vector_type(2))) float v2f;
typedef __attribute__((ext_vector_type(8))) float v8f;

// ---------------- kernel 1: per-box softmax stats (max, sum) ----------------
// One wave32 per box: lanes cooperatively read the 81 logits (coalesced),
// shuffle-reduce max and sum(exp(x-max)). 534MB single pass -> ~23us floor.
__global__ void softmax_stats_kernel(const float* __restrict__ p,
                                     float* __restrict__ smax,
                                     float* __restrict__ ssum,
                                     int totalBoxes) {
    int gtid = blockIdx.x * blockDim.x + threadIdx.x;
    int box  = gtid >> 5;
    int lane = gtid & 31;
    if (box >= totalBoxes) return;
    const float* row = p + (size_t)box * ROW + 4;   // 81 class logits
    float a = row[lane];            // idx 4..35
    float b = row[lane + 32];       // idx 36..67
    float c = (lane < 17) ? row[lane + 64] : -1e30f; // idx 68..84
    float m = fmaxf(a, fmaxf(b, c));
    #pragma unroll
    for (int off = 16; off > 0; off >>= 1) m = fmaxf(m, __shfl_xor(m, off, 32));
    float s = expf(a - m) + expf(b - m) + ((lane < 17) ? expf(c - m) : 0.0f);
    #pragma unroll
    for (int off = 16; off > 0; off >>= 1) s += __shfl_xor(s, off, 32);
    if (lane == 0) {
        smax[box] = m;
        ssum[box] = s;
    }
}

// ---------------- kernel 2: per-(batch,class) topk + NMS ----------------
struct SharedBuf {
    float               scores[NPAD];        // 98304 B : thresholded softmax scores
    unsigned long long  keys[CANDCAP];       //  4096 B : (score_bits<<32)|~idx
    float4              cor[KPAD];           //  3328 B : x1,y1,x2,y2
    float4              cent[KPAD];          //  3328 B : cx,cy,w,h (output form)
    float               area[KPAD];          //   832 B
    float               score[KPAD];         //   832 B
    float               keepf[KPAD];         //   832 B
    unsigned short      iou[KPAD][14];       //  5824 B : 208 x (13+1 pad) halfwords
    unsigned            hist[256];           //  1024 B : radix-select histogram
    unsigned            selPref;
    unsigned            selK;
    int                 cnt;
};
static_assert(sizeof(SharedBuf) < 160 * 1024, "must allow 2 WGs per 320KB WGP");

__global__ void select_nms_kernel(const float* __restrict__ p,
                                  const float* __restrict__ smax,
                                  const float* __restrict__ ssum,
                                  float* __restrict__ out) {
    extern __shared__ unsigned char k2_smem[];
    SharedBuf& sb = *reinterpret_cast<SharedBuf*>(k2_smem);

    const int tid  = threadIdx.x;
    const int inst = blockIdx.x;             // 0..5119
    const int b    = inst / NCLS;
    const int cls  = inst - b * NCLS;
    const float* bbase = p + (size_t)b * NBOX * ROW;
    const float* mrow  = smax + (size_t)b * NBOX;
    const float* srow  = ssum + (size_t)b * NBOX;

    // ---- stage thresholded scores for this class column into LDS ----
    for (int n = tid; n < NPAD; n += THREADS) {
        float sc = 0.0f;
        if (n < NBOX) {
            if (n + THREADS < NBOX)   // gfx1250 global_prefetch_b8 for next strided row
                __builtin_prefetch(bbase + (size_t)(n + THREADS) * ROW + 4 + cls, 0, 1);
            float l = bbase[(size_t)n * ROW + 4 + cls];
            float v = expf(l - mrow[n]) / srow[n];     // softmax score
            sc = (v > CONF_THR) ? v : 0.0f;            // mask (reference semantics)
        }
        sb.scores[n] = sc;
    }
    __syncthreads();

    // ---- MSB-first radix select: exact 200th-largest score value -----------
    // All passing scores lie in (0.01, 1]  =>  float bits[31:26] == 0b001111.
    // Digits: bits[25:18], [17:10], [9:2], [7:0] (last overlaps fixed bits,
    // which is harmless). 4 histogram passes instead of ~26 bisection passes.
    unsigned pref      = 0x3C000000u;
    unsigned fixedMask = 0xFC000000u;
    int      k         = TOPK;
    bool     few       = false;       // fewer than TOPK candidates
    const int shifts[4] = {18, 10, 2, 0};
    #pragma unroll 1
    for (int rr = 0; rr < 4; ++rr) {
        if (few) break;
        const int shift = shifts[rr];
        for (int i = tid; i < 256; i += THREADS) sb.hist[i] = 0u;
        __syncthreads();
        for (int n = tid; n < NBOX; n += THREADS) {
            unsigned bits = __float_as_uint(sb.scores[n]);
            if (bits != 0u && (bits & fixedMask) == pref)
                atomicAdd(&sb.hist[(bits >> shift) & 0xFFu], 1u);
        }
        __syncthreads();
        if (tid == 0) {
            unsigned cum = 0u; int bsel = -1;
            for (int bin = 255; bin >= 0; --bin) {
                unsigned c = sb.hist[bin];
                if (cum + c >= (unsigned)k) { bsel = bin; break; }
                cum += c;
            }
            if (bsel < 0) { sb.selPref = 0xFFFFFFFFu; sb.selK = 0u; } // < k candidates
            else {
                sb.selPref = pref | ((unsigned)bsel << shift);
                sb.selK    = (unsigned)k - cum;
            }
        }
        __syncthreads();
        unsigned sp = sb.selPref;
        unsigned sk = sb.selK;
        __syncthreads();
        if (sp == 0xFFFFFFFFu) few = true;
        else { pref = sp; fixedMask |= (0xFFu << shift); k = (int)sk; }
    }
    // t: count(>t) < TOPK <= count(>=t)   (or all candidates when few)
    const unsigned t = few ? (__float_as_uint(CONF_THR) + 1u) : pref;

    // ---- compact candidates (score_bits >= t) into key buffer ----
    for (int i = tid; i < CANDCAP; i += THREADS) sb.keys[i] = 0ull;
    if (tid == 0) sb.cnt = 0;
    __syncthreads();
    for (int n = tid; n < NBOX; n += THREADS) {
        unsigned bits = __float_as_uint(sb.scores[n]);
        if (bits >= t) {
            int slot = atomicAdd(&sb.cnt, 1);
            if (slot < CANDCAP)
                sb.keys[slot] = ((unsigned long long)bits << 32) | (unsigned)(~n);
        }
    }
    __syncthreads();

    // ---- bitonic sort (ascending) of 512 keys; top-r = keys[511-r].
    //      key = (score_bits, ~idx): descending read order == score desc, idx asc,
    //      matching lax.top_k tie-breaking. ----
    for (unsigned kk = 2; kk <= CANDCAP; kk <<= 1) {
        for (unsigned j = kk >> 1; j > 0; j >>= 1) {
            for (unsigned i = tid; i < CANDCAP; i += THREADS) {
                unsigned ixj = i ^ j;
                if (ixj > i) {
                    unsigned long long a = sb.keys[i];
                    unsigned long long c = sb.keys[ixj];
                    bool up = ((i & kk) == 0);
                    if (up ? (a > c) : (a < c)) { sb.keys[i] = c; sb.keys[ixj] = a; }
                }
            }
            __syncthreads();
        }
    }

    // ---- unpack top-200, gather boxes, build corners/areas (pad to 208) ----
    for (int r = tid; r < KPAD; r += THREADS) {
        if (r < TOPK) {
            unsigned long long key = sb.keys[CANDCAP - 1 - r];
            unsigned bits = (unsigned)(key >> 32);
            float sc = __uint_as_float(bits);
            bool valid = (sc > CONF_THR);
            unsigned idx = valid ? ~(unsigned)key : 0u;  // safe gather when padded
            const float* lp = bbase + (size_t)idx * ROW;
            float cx = lp[0], cy = lp[1], w = lp[2], h = lp[3];
            sb.cent[r]  = make_float4(cx, cy, w, h);
            float4 c4   = make_float4(cx - w * 0.5f, cy - h * 0.5f,
                                      cx + w * 0.5f, cy + h * 0.5f);
            sb.cor[r]   = c4;
            sb.area[r]  = (c4.z - c4.x) * (c4.w - c4.y);
            sb.score[r] = sc;
        } else {                                   // WMMA padding rows: no overlap
            sb.cent[r]  = make_float4(0.f, 0.f, 0.f, 0.f);
            sb.cor[r]   = make_float4(1e30f, 1e30f, 1e30f, 1e30f);
            sb.area[r]  = 0.0f;
            sb.score[r] = 0.0f;
        }
        sb.iou[r][13] = 0;      // pad halfword so NMS can read 7 full u32 words
    }
    __syncthreads();

    // ---- IoU bit-matrix via WMMA tiles -------------------------------------
    // union base U[i][j] = area_i + area_j is the rank-2 product
    //   [area_i, 1] x [1, area_j]^T  ->  one V_WMMA_F32_16X16X4_F32 per 16x16 tile
    // (f32 in/out, RNE accumulation => bit-identical to scalar adds).
    // Each of the 8 accumulator rows is converted to 32 predicate bits with one
    // wave32 ballot (lanes 0-15 -> row mB+v, lanes 16-31 -> row mB+8+v); each
    // (row, tile-column) halfword has a single writer => plain b16 store,
    // no atomics and no clear pass.
    {
        const int wv    = tid >> 5;
        const int lane  = tid & 31;
        const int laneM = lane & 15;
        const bool lo16 = (lane < 16);
        for (int tile = wv; tile < NTILE * NTILE; tile += (THREADS / 32)) {
            int ti = tile / NTILE, tj = tile - ti * NTILE;
            int mB = ti * 16, nB = tj * 16;
            // A 16x4 layout: lanes 0-15 -> K=0,1 ; lanes 16-31 -> K=2,3 (zero pad)
            v2f av, bv;
            av.x = lo16 ? sb.area[mB + laneM] : 0.0f;   // K=0 column: area_i
            av.y = lo16 ? 1.0f : 0.0f;                  // K=1 column: 1
            bv.x = lo16 ? 1.0f : 0.0f;                  // K=0 row: 1
            bv.y = lo16 ? sb.area[nB + laneM] : 0.0f;   // K=1 row: area_j
            v8f acc = {0.f, 0.f, 0.f, 0.f, 0.f, 0.f, 0.f, 0.f};
            acc = __builtin_amdgcn_wmma_f32_16x16x4_f32(
                      false, av, false, bv, (short)0, acc, false, false);
            // C layout: lane<16 -> M = mB+v, lane>=16 -> M = mB+8+v ; N = nB+laneM
            int nCol = nB + laneM;
            float4 cj = sb.cor[nCol];
            int mOff = mB + (lo16 ? 0 : 8);
            #pragma unroll
            for (int v = 0; v < 8; ++v) {
                float4 ci = sb.cor[mOff + v];
                float ww = fmaxf(fminf(ci.z, cj.z) - fmaxf(ci.x, cj.x), 0.0f);
                float hh = fmaxf(fminf(ci.w, cj.w) - fmaxf(ci.y, cj.y), 0.0f);
                float inter = ww * hh;
                float uni   = acc[v] - inter;            // area_i+area_j from WMMA
                // iou > thr  <=>  inter > thr * max(uni, 1e-8)   (no divide)
                bool pred = inter > IOU_THR * fmaxf(uni, 1e-8f);
                unsigned bal = (unsigned)__ballot(pred);
                if (lane == 0)
                    sb.iou[mB + v][tj]     = (unsigned short)(bal & 0xFFFFu);
                else if (lane == 16)
                    sb.iou[mB + 8 + v][tj] = (unsigned short)(bal >> 16);
            }
        }
    }
    __syncthreads();

    // ---- sequential greedy NMS scan: 200 steps of 7-word AND (thread 0) ----
    if (tid == 0) {
        unsigned kept[7] = {0u, 0u, 0u, 0u, 0u, 0u, 0u};
        for (int i = 0; i < TOPK; ++i) {
            const unsigned* rowp = (const unsigned*)sb.iou[i];  // 28B row, 4-aligned
            unsigned sup = 0u;
            #pragma unroll
            for (int w = 0; w < 7; ++w) sup |= (rowp[w] & kept[w]);
            bool kp = (sb.score[i] > CONF_THR) && (sup == 0u);
            sb.keepf[i] = kp ? 1.0f : 0.0f;
            if (kp) kept[i >> 5] |= (1u << (i & 31));
        }
    }
    __syncthreads();

    // ---- pack output: (B, 80, 200, 7) = keep, cls, score, cx, cy, w, h ----
    float* obase = out + (size_t)inst * TOPK * 7;
    const float fcls = (float)cls;
    for (int e = tid; e < TOPK * 7; e += THREADS) {
        int r = e / 7, f = e - r * 7;
        float val;
        if      (f == 0) val = sb.keepf[r];
        else if (f == 1) val = fcls;
        else if (f == 2) val = sb.score[r];
        else {
            float4 cc = sb.cent[r];
            val = (f == 3) ? cc.x : (f == 4) ? cc.y : (f == 5) ? cc.z : cc.w;
        }
        obase[e] = val;
    }
}

// ---------------- host launcher ----------------
extern "C" void kernel_launch(void* const* d_in, const int* in_sizes, int n_in,
                              void* d_out, int out_size, void* d_ws, size_t ws_size,
                              hipStream_t stream) {
    (void)in_sizes; (void)n_in; (void)out_size; (void)ws_size;
    const float* predicts = (const float*)d_in[0];
    float* out  = (float*)d_out;
    float* smax = (float*)d_ws;                       // B*N floats
    float* ssum = smax + (size_t)BATCH * NBOX;        // B*N floats (12.6 MB total)

    const int totalBoxes = BATCH * NBOX;              // 1,572,096
    // kernel 1: one wave32 per box
    {
        long long threadsNeeded = (long long)totalBoxes * 32;
        int blocks = (int)((threadsNeeded + THREADS - 1) / THREADS);
        softmax_stats_kernel<<<blocks, THREADS, 0, stream>>>(
            predicts, smax, ssum, totalBoxes);
    }
    // kernel 2: one workgroup per (batch, class) instance, ~119KB dynamic LDS
    {
        select_nms_kernel<<<BATCH * NCLS, THREADS, sizeof(SharedBuf), stream>>>(
            predicts, smax, ssum, out);
    }
}